// TophatIntegrator_16930761081440
// MI455X (gfx1250) — compile-verified
//
#include <hip/hip_runtime.h>
#include <stdint.h>

// Problem constants (from reference)
#define ROWS     128
#define TLEN     524288
#define WSZ      2049                      // window taps
#define TILE     16384                     // outputs per block
#define CHUNKS   (TLEN / TILE)             // 32 blocks per row
#define NTHREADS 256
#define REGION   (TILE + WSZ - 1)          // 18432 floats staged in LDS (72 KB)
#define CPT      (REGION / NTHREADS)       // 72 elements scanned per thread
#define NITER    (REGION / (NTHREADS * 4)) // 18 async b128 loads per thread

typedef float v8f __attribute__((ext_vector_type(8)));
typedef float v2f __attribute__((ext_vector_type(2)));

__global__ __launch_bounds__(NTHREADS)
void tophat_kernel(const float* __restrict__ x, float* __restrict__ out) {
  __shared__ __align__(16) float xs[REGION];  // staged region, then in-place cumsum
  __shared__ float tsum[NTHREADS];            // per-thread chunk sums
  __shared__ float rsum[16];                  // row totals for WMMA scan
  __shared__ float excl[NTHREADS];            // exclusive prefix of tsum

  const int tid   = threadIdx.x;
  const int b     = blockIdx.x;
  const int row   = b / CHUNKS;
  const int chunk = b % CHUNKS;
  const int t0    = chunk * TILE;
  const int start = t0 - (WSZ - 1);           // first region element (may be < 0 for chunk 0)
  const float* xrow = x + (size_t)row * TLEN;

  // ---- Phase 1: stage REGION floats into LDS ----
  if (chunk == 0) {
    // Left edge: zero-pad negative indices with plain guarded loads.
    for (int i = tid; i < REGION; i += NTHREADS) {
      int g = start + i;
      xs[i] = (g >= 0) ? xrow[g] : 0.0f;
    }
    __syncthreads();
  } else {
    // CDNA5 async copy: global -> LDS, 16B per lane per issue, tracked by ASYNCcnt.
    uint64_t gbase = (uint64_t)(uintptr_t)(xrow + start);      // 16B aligned (start % 2048 == 0)
    uint32_t lbase = (uint32_t)(uintptr_t)(void*)&xs[0];       // LDS byte offset
#pragma unroll
    for (int k = 0; k < NITER; ++k) {
      uint32_t boff = (uint32_t)(k * NTHREADS + tid) * 16u;
      uint32_t ldst = lbase + boff;
      asm volatile("global_load_async_to_lds_b128 %0, %1, %2"
                   :: "v"(ldst), "v"(boff), "s"(gbase) : "memory");
    }
    asm volatile("s_wait_asynccnt 0" ::: "memory");
    __syncthreads();
  }

  // ---- Phase 2: per-thread sequential scan of CPT contiguous elements ----
  float vals[CPT];
  float acc = 0.0f;
  const int base = tid * CPT;
#pragma unroll
  for (int i = 0; i < CPT; ++i) { acc += xs[base + i]; vals[i] = acc; }
  tsum[tid] = acc;
  __syncthreads();

  // ---- Phase 3: exclusive scan of the 256 thread sums ----
  float offset;
#if __has_builtin(__builtin_amdgcn_wmma_f32_16x16x4_f32)
  {
    // View tsum as S[i][j] = tsum[16i+j]. P = S*U (row inclusive prefix),
    // R = Lstrict * rowtotals, excl = R + P - S. Every wave computes the
    // identical result (EXEC all ones for WMMA); writes race benignly.
    const int lane = tid & 31;
    const int n    = lane & 15;   // column / M-row index for this lane
    const int hi   = lane >> 4;   // half-wave select (K offset +2, M offset +8)

    v8f p = {0.f, 0.f, 0.f, 0.f, 0.f, 0.f, 0.f, 0.f};
#pragma unroll
    for (int k = 0; k < 4; ++k) {
      const int kk0 = 4 * k + 2 * hi;
      v2f a, bm;
      a.x  = tsum[16 * n + kk0];              // A[m=n][kk0]
      a.y  = tsum[16 * n + kk0 + 1];          // A[m=n][kk0+1]
      bm.x = (kk0     <= n) ? 1.0f : 0.0f;    // U[kk0][n]
      bm.y = (kk0 + 1 <= n) ? 1.0f : 0.0f;    // U[kk0+1][n]
      p = __builtin_amdgcn_wmma_f32_16x16x4_f32(false, a, false, bm,
                                                (short)0, p, false, false);
    }
    // Row totals r[i] = P[i][15] live in lanes 15 (rows 0..7) and 31 (rows 8..15).
    if (n == 15) {
#pragma unroll
      for (int v = 0; v < 8; ++v) rsum[8 * hi + v] = p[v];
    }
    v8f rm = {0.f, 0.f, 0.f, 0.f, 0.f, 0.f, 0.f, 0.f};
#pragma unroll
    for (int k = 0; k < 4; ++k) {
      const int kk0 = 4 * k + 2 * hi;
      v2f a, bm;
      a.x  = (kk0     < n) ? 1.0f : 0.0f;     // Lstrict[m=n][kk0]
      a.y  = (kk0 + 1 < n) ? 1.0f : 0.0f;     // Lstrict[m=n][kk0+1]
      bm.x = rsum[kk0];                       // broadcast r across columns
      bm.y = rsum[kk0 + 1];
      rm = __builtin_amdgcn_wmma_f32_16x16x4_f32(false, a, false, bm,
                                                 (short)0, rm, false, false);
    }
#pragma unroll
    for (int v = 0; v < 8; ++v) {
      const int idx = 16 * (v + 8 * hi) + n;  // (row, col) -> flat
      excl[idx] = rm[v] + p[v] - tsum[idx];   // exclusive = rowoff + incl - self
    }
  }
  __syncthreads();
  offset = excl[tid];
#else
  // Fallback: Hillis-Steele inclusive scan in LDS.
  float myv = tsum[tid];
  for (int off = 1; off < NTHREADS; off <<= 1) {
    float add = (tid >= off) ? tsum[tid - off] : 0.0f;
    __syncthreads();
    myv += add;
    tsum[tid] = myv;
    __syncthreads();
  }
  offset = myv - acc;
#endif

  // ---- Phase 4: write back globally-offset cumsum for this region ----
#pragma unroll
  for (int i = 0; i < CPT; ++i) xs[base + i] = vals[i] + offset;
  __syncthreads();

  // ---- Phase 5: windowed difference, coalesced stores ----
  const float invW = 1.0f / (float)WSZ;
  float* orow = out + (size_t)row * TLEN + t0;
#pragma unroll 4
  for (int i = 0; i < TILE / NTHREADS; ++i) {
    const int j   = i * NTHREADS + tid;
    const float h = xs[j + (WSZ - 1)];
    const float l = (j > 0) ? xs[j - 1] : 0.0f;
    orow[j] = (h - l) * invW;
  }
}

extern "C" void kernel_launch(void* const* d_in, const int* in_sizes, int n_in,
                              void* d_out, int out_size, void* d_ws, size_t ws_size,
                              hipStream_t stream) {
  (void)in_sizes; (void)n_in; (void)out_size; (void)d_ws; (void)ws_size;
  const float* x = (const float*)d_in[0];
  float* out = (float*)d_out;
  dim3 grid(ROWS * CHUNKS);   // 4096 blocks
  dim3 blk(NTHREADS);         // 256 threads = 8 wave32
  hipLaunchKernelGGL(tophat_kernel, grid, blk, 0, stream, x, out);
}